// LeftDiagonalStripConv_54159537603007
// MI455X (gfx1250) — compile-verified
//
#include <hip/hip_runtime.h>
#include <math.h>
#include <stdint.h>

// ---------------------------------------------------------------------------
// Problem constants (match reference)
// ---------------------------------------------------------------------------
#define B_ 8
#define C_ 64
#define H_ 256
#define W_ 256

constexpr int TILE   = 30;              // output tile (per workgroup)
constexpr int NTIL   = 9;               // ceil(256/30)
constexpr int XSROWS = 32;              // TILE + 2 (conv halo)
constexpr int XSCOLS = 64;              // skew-diagonal window (63 used)
constexpr int XSTR   = 66;              // xs LDS row stride (bank pad)
constexpr int UROWS  = 52;              // XSROWS + 2*9 + 2 (K padding)
constexpr int UCOLS  = 64;
constexpr int USTR   = 80;              // u LDS row stride: 16B-aligned rows,
                                        // +16 bank shift per row -> conflict-free B loads
constexpr int BLOCKS_PER_CH = B_ * NTIL * NTIL;   // 648
constexpr float INV_NPIX = 1.0f / (float)(B_ * H_ * W_);   // 1/524288

typedef float v2f __attribute__((ext_vector_type(2)));
typedef float v8f __attribute__((ext_vector_type(8)));

// ---------------------------------------------------------------------------
// CDNA5 feature detection (device pass only; host pass compiles fallbacks)
// ---------------------------------------------------------------------------
#if defined(__gfx1250__) && __has_builtin(__builtin_amdgcn_wmma_f32_16x16x4_f32)
#define USE_WMMA 1
#else
#define USE_WMMA 0
#endif

#if defined(__gfx1250__) && __has_builtin(__builtin_amdgcn_global_load_async_to_lds_b128)
#define USE_ASYNC 1
#else
#define USE_ASYNC 0
#endif

// Compile-time probes: fail loudly (device pass only) if a CDNA5 path is
// silently unavailable, so the compile log tells us which paths lowered.
#if defined(__gfx1250__) && !USE_WMMA
#error "gfx1250 device pass: __builtin_amdgcn_wmma_f32_16x16x4_f32 unavailable"
#endif
#if defined(__gfx1250__) && !USE_ASYNC
#error "gfx1250 device pass: __builtin_amdgcn_global_load_async_to_lds_b128 unavailable"
#endif

#if USE_ASYNC
// Builtin prototype is (v4i AS1*, v4i AS3*, imm int, imm int) where v4i is a
// GCC-style vector_size(16) int ("__device__" in HIP diagnostics == AS1).
typedef int v4i_b __attribute__((vector_size(16)));
typedef v4i_b __attribute__((address_space(1))) v4i_glb;
typedef v4i_b __attribute__((address_space(3))) v4i_lds;
#endif

__device__ __forceinline__ void wait_async_lds() {
#if defined(__gfx1250__)
#if __has_builtin(__builtin_amdgcn_s_wait_asynccnt)
  __builtin_amdgcn_s_wait_asynccnt(0);
#else
  asm volatile("s_wait_asynccnt 0" ::: "memory");
#endif
#endif
}

// ---------------------------------------------------------------------------
// Fused strip-sum + dwconv3x3 core.
// STATS=true : accumulate per-block partial sum / sumsq of y into workspace.
// STATS=false: apply per-channel scale/shift + exact GELU, write output.
// ---------------------------------------------------------------------------
template <bool STATS>
__global__ __launch_bounds__(256)
void strip_conv_kernel(const float* __restrict__ x,
                       const float* __restrict__ dwk,
                       float* __restrict__ pSum,
                       float* __restrict__ pSq,
                       const float* __restrict__ scaleC,
                       const float* __restrict__ shiftC,
                       float* __restrict__ out)
{
  __shared__ float u[UROWS * USTR];     // skewed x tile: u[r][c] = x(r0+r, c + r + colbase)
  __shared__ float xs[XSROWS * XSTR];   // strip sums in skewed coords
  __shared__ float red[512];

  const int tid  = threadIdx.x;
  const int tile = blockIdx.x % (NTIL * NTIL);
  const int bc   = blockIdx.x / (NTIL * NTIL);
  const int c    = bc % C_;
  const int b    = bc / C_;
  const int ty   = tile / NTIL;
  const int tx   = tile % NTIL;
  const int h0   = ty * TILE;
  const int w0   = tx * TILE;

  const float* xc = x + (size_t)(b * C_ + c) * (H_ * W_);

  // -------------------------------------------------------------------
  // Stage 1: load skewed tile u[ur][uc] = x(h0-10+ur, w0-41+ur+uc), 0 if OOB
  // -------------------------------------------------------------------
  const bool interior = (h0 - 10 >= 0) && (h0 + 41 < H_) &&
                        (w0 - 41 >= 0) && (w0 + 73 < W_);
  if (interior) {
#if USE_ASYNC
    // Per-lane independent LDS destinations: async b128, 4 floats per lane.
    for (int q = tid; q < UROWS * (UCOLS / 4); q += 256) {
      const int ur = q >> 4;
      const int uq = (q & 15) << 2;
      const float* gp = xc + (h0 - 10 + ur) * W_ + (w0 - 41 + ur + uq);
      const uint32_t loff = (uint32_t)(uintptr_t)(&u[ur * USTR + uq]);
      __builtin_amdgcn_global_load_async_to_lds_b128(
          (v4i_glb*)(uintptr_t)gp, (v4i_lds*)(uintptr_t)loff, 0, 0);
    }
    wait_async_lds();
#else
    for (int q = tid; q < UROWS * (UCOLS / 4); q += 256) {
      const int ur = q >> 4;
      const int uq = (q & 15) << 2;
      const float* gp = xc + (h0 - 10 + ur) * W_ + (w0 - 41 + ur + uq);
      float4 v = *(const float4*)gp;
      *(float4*)(&u[ur * USTR + uq]) = v;
    }
#endif
  } else {
    // Boundary tiles: guarded scalar loads implement zero padding.
    for (int e = tid; e < UROWS * UCOLS; e += 256) {   // 3328 = 13*256, no divergence
      const int ur = e >> 6;
      const int uc = e & 63;
      const int gr = h0 - 10 + ur;
      const int gc = w0 - 41 + ur + uc;
      float v = 0.0f;
      if (gr >= 0 && gr < H_ && gc >= 0 && gc < W_) v = xc[gr * W_ + gc];
      u[ur * USTR + uc] = v;
    }
  }
  __syncthreads();

  // -------------------------------------------------------------------
  // Stage 2: 19-tap diagonal strip sum == banded matmul in skewed coords.
  // xs[s][cl] = (1/19) * sum_{i=0..18} u[s+i][cl]
  // One 16x16 output tile per wave (8 waves = 2 row-blocks x 4 col-blocks).
  // -------------------------------------------------------------------
#if USE_WMMA
  {
    const int lane = tid & 31;
    const int wv   = tid >> 5;
    const int hb   = (wv >> 2) << 4;    // 0 or 16
    const int cbl  = (wv & 3) << 4;     // 0,16,32,48
    const int m    = lane & 15;
    const int kh   = (lane & 16) ? 2 : 0;   // K split across lane halves
    v8f acc = {};
#pragma unroll
    for (int j = 0; j < 9; ++j) {       // K = 4 per step, 36 input rows (band-masked)
      const int k0 = 4 * j + kh;
      v2f a, bb;
      // A[m][k] = 1/19 iff m <= k <= m+18 (output row m sums input rows m..m+18)
      a.x = (k0     >= m && k0     <= m + 18) ? (1.0f / 19.0f) : 0.0f;
      a.y = (k0 + 1 >= m && k0 + 1 <= m + 18) ? (1.0f / 19.0f) : 0.0f;
      const int rb = hb + k0;
      bb.x = u[(rb    ) * USTR + cbl + m];
      bb.y = u[(rb + 1) * USTR + cbl + m];
      acc = __builtin_amdgcn_wmma_f32_16x16x4_f32(
          false, a, false, bb, (short)0, acc, false, false);
    }
#pragma unroll
    for (int v = 0; v < 8; ++v) {       // C/D layout: VGPR v <-> M = v / v+8
      const int mm = v + ((lane >> 4) << 3);
      xs[(hb + mm) * XSTR + cbl + m] = acc[v];
    }
  }
#else
  for (int e = tid; e < XSROWS * XSCOLS; e += 256) {
    const int s  = e >> 6;
    const int cl = e & 63;
    float t = 0.0f;
#pragma unroll
    for (int i = 0; i < 19; ++i) t += u[(s + i) * USTR + cl];
    xs[s * XSTR + cl] = t * (1.0f / 19.0f);
  }
#endif
  __syncthreads();

  // -------------------------------------------------------------------
  // Stage 3: depthwise 3x3 conv (zero-padded on xs) + epilogue
  // -------------------------------------------------------------------
  float kw[9];
#pragma unroll
  for (int i = 0; i < 9; ++i) kw[i] = dwk[c * 9 + i];

  float sc = 0.0f, sh = 0.0f;
  if (!STATS) { sc = scaleC[c]; sh = shiftC[c]; }

  float lsum = 0.0f, lsq = 0.0f;
  for (int e = tid; e < TILE * TILE; e += 256) {
    const int oy = e / TILE;
    const int ox = e - oy * TILE;
    const int h = h0 + oy;
    const int w = w0 + ox;
    if (h < H_ && w < W_) {
      float y = 0.0f;
#pragma unroll
      for (int dh = -1; dh <= 1; ++dh) {
#pragma unroll
        for (int dw = -1; dw <= 1; ++dw) {
          const int hh = h + dh;
          const int ww = w + dw;
          float v = 0.0f;   // conv reads 0 outside the image (xs zero-pad)
          if (hh >= 0 && hh < H_ && ww >= 0 && ww < W_)
            v = xs[(oy + 1 + dh) * XSTR + (ox - oy + 31 + dw - dh)];
          y = fmaf(kw[(dh + 1) * 3 + (dw + 1)], v, y);
        }
      }
      if (STATS) {
        lsum += y;
        lsq = fmaf(y, y, lsq);
      } else {
        const float yn = fmaf(y, sc, sh);
        out[((size_t)(b * C_ + c) * H_ + h) * W_ + w] =
            0.5f * yn * (1.0f + erff(yn * 0.70710678118654752f));
      }
    }
  }

  if (STATS) {
    red[tid]       = lsum;
    red[256 + tid] = lsq;
    __syncthreads();
    for (int off = 128; off > 0; off >>= 1) {
      if (tid < off) {
        red[tid]       += red[tid + off];
        red[256 + tid] += red[256 + tid + off];
      }
      __syncthreads();
    }
    if (tid == 0) {
      const int slot = b * (NTIL * NTIL) + tile;
      pSum[c * BLOCKS_PER_CH + slot] = red[0];
      pSq [c * BLOCKS_PER_CH + slot] = red[256];
    }
  }
}

// ---------------------------------------------------------------------------
// Per-channel BN finalize: deterministic fixed-order reduction of partials.
// scale = gamma*rsqrt(var+eps), shift = beta - mean*scale
// ---------------------------------------------------------------------------
__global__ void bn_finalize_kernel(const float* __restrict__ pSum,
                                   const float* __restrict__ pSq,
                                   const float* __restrict__ gamma,
                                   const float* __restrict__ beta,
                                   float* __restrict__ scaleC,
                                   float* __restrict__ shiftC)
{
  const int c = threadIdx.x;
  if (c < C_) {
    float s = 0.0f, q = 0.0f;
    for (int i = 0; i < BLOCKS_PER_CH; ++i) {
      s += pSum[c * BLOCKS_PER_CH + i];
      q += pSq [c * BLOCKS_PER_CH + i];
    }
    const float mean = s * INV_NPIX;
    const float var  = q * INV_NPIX - mean * mean;
    const float rstd = rsqrtf(var + 1e-5f);
    const float sc   = gamma[c] * rstd;
    scaleC[c] = sc;
    shiftC[c] = beta[c] - mean * sc;
  }
}

// ---------------------------------------------------------------------------
// Entry point
// ---------------------------------------------------------------------------
extern "C" void kernel_launch(void* const* d_in, const int* in_sizes, int n_in,
                              void* d_out, int out_size, void* d_ws, size_t ws_size,
                              hipStream_t stream) {
  const float* x     = (const float*)d_in[0];
  const float* dwk   = (const float*)d_in[1];
  const float* gamma = (const float*)d_in[2];
  const float* beta  = (const float*)d_in[3];
  float* out = (float*)d_out;

  // Workspace layout (floats): partials (64*648 each) + scale/shift (64 each)
  float* ws   = (float*)d_ws;
  const int P = C_ * BLOCKS_PER_CH;     // 41472
  float* pSum   = ws;
  float* pSq    = ws + P;
  float* scaleC = ws + 2 * P;
  float* shiftC = ws + 2 * P + C_;
  // total: 2*41472 + 128 floats = ~332 KB

  const dim3 grid(B_ * C_ * NTIL * NTIL);   // 41472 workgroups
  const dim3 blk(256);

  strip_conv_kernel<true><<<grid, blk, 0, stream>>>(
      x, dwk, pSum, pSq, nullptr, nullptr, nullptr);
  bn_finalize_kernel<<<dim3(1), dim3(64), 0, stream>>>(
      pSum, pSq, gamma, beta, scaleC, shiftC);
  strip_conv_kernel<false><<<grid, blk, 0, stream>>>(
      x, dwk, nullptr, nullptr, scaleC, shiftC, out);
}